// GroupPoints_9070970929656
// MI455X (gfx1250) — compile-verified
//
#include <hip/hip_runtime.h>
#include <hip/hip_bf16.h>
#include <math.h>

typedef __attribute__((ext_vector_type(2))) float v2f;
typedef __attribute__((ext_vector_type(8))) float v8f;

#define B_    16
#define NS    4096
#define NT    1024
#define KK    32
#define TILES (NS / 16)

// output layout (floats, concatenated in reference return order)
#define OFF_PATCHES 0L          // (16,1024,32,3) = 1572864
#define OFF_PIDX    1572864L    // (16,1024,32)   = 524288
#define OFF_PSZ     2097152L    // (16,1024)      = 16384
#define OFF_RAD     2113536L    // (16,1,1)       = 16
#define OFF_PDIST   2113552L    // (16,1024,32)   = 524288

__global__ __launch_bounds__(32)
void group_points_topk_wmma(const float* __restrict__ src,
                            const float* __restrict__ tgt,
                            float* __restrict__ out)
{
    // per-lane top-32 half-lists staged for the final pairwise merge
    __shared__ float    s_d[32 * 33];   // 33-stride padding keeps p==32 reads in-bounds
    __shared__ unsigned s_x[32 * 33];

    const int  lane  = threadIdx.x;     // 0..31 (wave32)
    const int  blk   = blockIdx.x;      // 0..1023
    const int  b     = blk >> 6;        // batch
    const int  tileT = blk & 63;        // 16-target tile
    const int  col   = lane & 15;       // target column this lane owns (N)
    const bool hi    = lane >= 16;      // upper half supplies K=2,3 rows
    const int  half8 = hi ? 8 : 0;      // D rows v+8 for upper lanes

    // ---- this lane's target point (column N = col of the B matrix) ----
    const int    tIdx = tileT * 16 + col;
    const float* tp   = tgt + ((long)b * NT + tIdx) * 3;
    const float  tx = tp[0], ty = tp[1], tz = tp[2];
    const float  tt = tx * tx + ty * ty + tz * tz;

    // B (4x16): VGPR0 = K0 | K2 halves, VGPR1 = K1 | K3 halves
    v2f bmat;
    bmat.x = hi ? tz : tx;
    bmat.y = hi ? 1.0f : ty;

    // C: lane l holds column N=l%16 in every VGPR -> splat |t_N|^2
    v8f cvec;
#pragma unroll
    for (int v = 0; v < 8; ++v) cvec[v] = tt;

    // ---- per-lane sorted (ascending) top-32 over this lane's source half ----
    float    ld[KK];
    unsigned lx[KK];
#pragma unroll
    for (int i = 0; i < KK; ++i) { ld[i] = __builtin_inff(); lx[i] = 0u; }

    const float* srcB = src + (long)b * NS * 3;

    // software pipeline: preload tile 0's source point
    float sx = srcB[(long)col * 3 + 0];
    float sy = srcB[(long)col * 3 + 1];
    float sz = srcB[(long)col * 3 + 2];

#pragma unroll 1
    for (int n = 0; n < TILES; ++n) {
        // A (16x4, sources as rows): lane supplies row M = col of this tile
        v2f amat;
        amat.x = hi ? (-2.0f * sz) : (-2.0f * sx);
        amat.y = hi ? (sx * sx + sy * sy + sz * sz) : (-2.0f * sy);

        // unconditional wrapped preload of next tile's point -> the s_wait can
        // sink past the WMMA and insert block instead of stalling immediately
        const int    nn = (n + 1) & (TILES - 1);
        const float* np = srcB + (long)(nn * 16 + col) * 3;
        const float nsx = np[0], nsy = np[1], nsz = np[2];

        // D = |t|^2 - 2 s.t + |s|^2  (exact squared distance, one WMMA per tile)
        v8f acc = __builtin_amdgcn_wmma_f32_16x16x4_f32(
            false, amat, false, bmat, (short)0, cvec, false, false);

        // tile-level prune: single guard for all 8 candidates
        float tmin = fminf(acc[0], acc[1]);
        tmin = fminf(tmin, fminf(acc[2], acc[3]));
        tmin = fminf(tmin, fminf(acc[4], acc[5]));
        tmin = fminf(tmin, fminf(acc[6], acc[7]));

        if (tmin < ld[KK - 1]) {
#pragma unroll
            for (int v = 0; v < 8; ++v) {
                const float    c  = acc[v];
                const unsigned ci = (unsigned)(n * 16 + v + half8);
                if (c < ld[KK - 1]) {
                    // branch-free shift-insert into ascending list; one v_cmp
                    // per position (shared: mpl_i == msh_{i+1}), all data moves
                    // as bit-blends -> v_cndmask pairs, no canonicalize ops
                    const unsigned cb = __builtin_bit_cast(unsigned, c);
                    unsigned mpl = 0xFFFFFFFFu;   // c < ld[KK-1] holds here
#pragma unroll
                    for (int i = KK - 1; i >= 1; --i) {
                        const unsigned msh = (c < ld[i - 1]) ? 0xFFFFFFFFu : 0u;
                        const unsigned dp  = __builtin_bit_cast(unsigned, ld[i - 1]);
                        const unsigned dc  = __builtin_bit_cast(unsigned, ld[i]);
                        const unsigned tdv = (cb & mpl) | (dc & ~mpl);
                        const unsigned tdi = (ci & mpl) | (lx[i] & ~mpl);
                        ld[i] = __builtin_bit_cast(float, (dp & msh) | (tdv & ~msh));
                        lx[i] = (lx[i - 1] & msh) | (tdi & ~msh);
                        mpl = msh;
                    }
                    const unsigned d0 = __builtin_bit_cast(unsigned, ld[0]);
                    ld[0] = __builtin_bit_cast(float, (cb & mpl) | (d0 & ~mpl));
                    lx[0] = (ci & mpl) | (lx[0] & ~mpl);
                }
            }
        }

        sx = nsx; sy = nsy; sz = nsz;
    }

    // ---- merge the two half-lists of each lane pair (l, l+16) via LDS ----
#pragma unroll
    for (int i = 0; i < KK; ++i) {
        s_d[lane * 33 + i] = ld[i];
        s_x[lane * 33 + i] = lx[i];
    }
    __syncthreads();

    if (lane < 16) {
        float* patches = out + OFF_PATCHES;
        float* pidx_o  = out + OFF_PIDX;
        float* psz_o   = out + OFF_PSZ;
        float* pdist_o = out + OFF_PDIST;

        const long tb    = (long)b * NT + tIdx;
        const int  base0 = lane * 33;
        const int  base1 = (lane + 16) * 33;
        int p0 = 0, p1 = 0;

        for (int k = 0; k < KK; ++k) {
            const float d0 = (p0 < KK) ? s_d[base0 + p0] : __builtin_inff();
            const float d1 = (p1 < KK) ? s_d[base1 + p1] : __builtin_inff();
            const bool  take0 = d0 <= d1;
            const float    dv = take0 ? d0 : d1;
            const unsigned xv = take0 ? s_x[base0 + p0] : s_x[base1 + p1];
            if (take0) ++p0; else ++p1;

            const bool keep = dv <= 4.0f;                 // rad^2 >= sq_pd, rad=2
            pidx_o[tb * KK + k]  = keep ? (float)xv : -1.0f;
            pdist_o[tb * KK + k] = sqrtf(fmaxf(dv, 1e-9f)) * 0.5f;

            // torch flat-gather wraparound for masked (-1) indices
            long gl = (long)b * NS + (keep ? (long)xv : -1L);
            if (gl < 0) gl += (long)B_ * NS;
            const float* gp = src + gl * 3;
            float*       pp = patches + (tb * KK + k) * 3;
            pp[0] = gp[0] * 0.5f - tx * 0.5f;
            pp[1] = gp[1] * 0.5f - ty * 0.5f;
            pp[2] = gp[2] * 0.5f - tz * 0.5f;
        }
        // gather of an all-ones mask is identically 1 -> sum over K == 32
        psz_o[tb] = 32.0f;
    }
    if (blockIdx.x == 0 && lane < 16) out[OFF_RAD + lane] = 2.0f;
}

extern "C" void kernel_launch(void* const* d_in, const int* in_sizes, int n_in,
                              void* d_out, int out_size, void* d_ws, size_t ws_size,
                              hipStream_t stream)
{
    const float* src = (const float*)d_in[0];   // (16, 4096, 3) fp32
    const float* tgt = (const float*)d_in[1];   // (16, 1024, 3) fp32
    float*       out = (float*)d_out;

    dim3 grid(B_ * (NT / 16));  // 1024 waves, one 16-target tile each
    dim3 block(32);             // one wave32
    group_points_topk_wmma<<<grid, block, 0, stream>>>(src, tgt, out);
}